// GINLayer_28209345200546
// MI455X (gfx1250) — compile-verified
//
#include <hip/hip_runtime.h>
#include <hip/hip_bf16.h>

#define NUM_NODES 10000
#define NUM_EDGES 640000
#define DIM       128

typedef __attribute__((ext_vector_type(2))) float v2f;
typedef __attribute__((ext_vector_type(8))) float v8f;

// ---------------------------------------------------------------------------
// Kernel 1: agg = x  (the "+ x" self-loop term of GIN; also initializes the
// accumulator for the scatter kernel). float4 copies -> global_load/store_b128.
// ---------------------------------------------------------------------------
__global__ void gin_init_agg(const float4* __restrict__ x,
                             float4* __restrict__ agg) {
    int i = blockIdx.x * blockDim.x + threadIdx.x;
    agg[i] = x[i];
}

// ---------------------------------------------------------------------------
// Kernel 2: agg[row[e]] += x[col[e]] — ONE WAVE32 PER EDGE.
// The edge id is wave-uniform, so edge_index is fetched via scalar memory
// (s_load_b64, KMcnt). Each lane covers 4 consecutive features:
// one coalesced global_load_b128 + 4 global_atomic_add_f32 into the
// L2-resident (5.12MB) agg buffer.
// ---------------------------------------------------------------------------
__global__ __launch_bounds__(256) void gin_edge_scatter(
    const float* __restrict__ x,
    const int* __restrict__ edge_index,   // (E,2) row-major
    float* __restrict__ agg) {
    const int wave = threadIdx.x >> 5;                  // 0..7
    const int lane = threadIdx.x & 31;
    const int e = blockIdx.x * 8 + wave;                // wave-uniform edge id

    const int r = edge_index[2 * e + 0];                // scalar loads
    const int c = edge_index[2 * e + 1];

    const int f0 = lane << 2;                           // 4 feats per lane
    const float4 v = *reinterpret_cast<const float4*>(x + (size_t)c * DIM + f0);
    float* dst = agg + (size_t)r * DIM + f0;
    atomicAdd(dst + 0, v.x);
    atomicAdd(dst + 1, v.y);
    atomicAdd(dst + 2, v.z);
    atomicAdd(dst + 3, v.w);
}

// ---------------------------------------------------------------------------
// Kernel 3: C = act(A @ W + b), A:[NUM_NODES,128], W:[128,128] (row-major),
// using V_WMMA_F32_16X16X4_F32. One wave32 per 16x16 output tile.
//
// VGPR layouts (CDNA5 ISA §7.12.2):
//   A 16x4 f32 : lane m=lane&15; lanes 0-15 hold K={0,1}, lanes 16-31 K={2,3}
//                (VGPR0=K even, VGPR1=K odd)
//   B 4x16 f32 : lane n=lane&15; lanes 0-15 hold K={0,1}, lanes 16-31 K={2,3}
//   C/D 16x16  : VGPR r, lanes 0-15 -> M=r, lanes 16-31 -> M=r+8, N=lane&15
// ---------------------------------------------------------------------------
__global__ __launch_bounds__(128) void gin_wmma_gemm(
    const float* __restrict__ A,
    const float* __restrict__ W,
    const float* __restrict__ bias,
    float* __restrict__ C,
    int do_relu) {
    const int wave = threadIdx.x >> 5;                 // 0..3
    const int lane = threadIdx.x & 31;
    const int tile = blockIdx.x * 4 + wave;            // 0..4999
    const int mt = tile >> 3;                          // row tile: 0..624
    const int nt = tile & 7;                           // col tile: 0..7

    const int half  = lane >> 4;                       // 0 = lanes 0-15, 1 = 16-31
    const int sub   = lane & 15;
    const int arow  = (mt << 4) + sub;                 // A row this lane feeds
    const int bcol  = (nt << 4) + sub;                 // W col this lane feeds
    const int koff  = half << 1;                       // 0 or 2

    v8f acc = {};
    for (int k = 0; k < DIM; k += 4) {
        v2f a, b;
        const float* ap = A + arow * DIM + (k + koff); // contiguous pair -> b64
        a.x = ap[0];
        a.y = ap[1];
        b.x = W[(k + koff + 0) * DIM + bcol];
        b.y = W[(k + koff + 1) * DIM + bcol];
        acc = __builtin_amdgcn_wmma_f32_16x16x4_f32(
            /*neg_a=*/false, a, /*neg_b=*/false, b,
            /*c_mod=*/(short)0, acc, /*reuse_a=*/false, /*reuse_b=*/false);
    }

    const int row0 = (mt << 4) + (half << 3);          // M base for this half-wave
    const float bv = bias[bcol];
#pragma unroll
    for (int r = 0; r < 8; ++r) {
        float v = acc[r] + bv;
        if (do_relu) v = fmaxf(v, 0.0f);
        C[(row0 + r) * DIM + bcol] = v;
    }
}

// ---------------------------------------------------------------------------
// Launch: init -> scatter -> GEMM1(relu) -> GEMM2
// Workspace: agg (5.12MB) + h (5.12MB) in d_ws.
// ---------------------------------------------------------------------------
extern "C" void kernel_launch(void* const* d_in, const int* in_sizes, int n_in,
                              void* d_out, int out_size, void* d_ws, size_t ws_size,
                              hipStream_t stream) {
    const float* x  = (const float*)d_in[0];
    const int*   ei = (const int*)d_in[1];
    const float* W1 = (const float*)d_in[2];
    const float* b1 = (const float*)d_in[3];
    const float* W2 = (const float*)d_in[4];
    const float* b2 = (const float*)d_in[5];
    float* out = (float*)d_out;

    float* agg = (float*)d_ws;                 // NUM_NODES*DIM floats
    float* h   = agg + (size_t)NUM_NODES * DIM;

    // 1) agg = x   (1.28M floats = 320K float4 elements)
    gin_init_agg<<<(NUM_NODES * DIM / 4) / 256, 256, 0, stream>>>(
        (const float4*)x, (float4*)agg);

    // 2) agg[row] += x[col]  (one wave per edge, 8 waves/block -> 80000 blocks)
    gin_edge_scatter<<<NUM_EDGES / 8, 256, 0, stream>>>(x, ei, agg);

    // 3) h = relu(agg @ W1 + b1) : 625*8 = 5000 tiles, 4 waves/block
    const int gemm_blocks = (NUM_NODES / 16) * (DIM / 16) / 4;
    gin_wmma_gemm<<<gemm_blocks, 128, 0, stream>>>(agg, W1, b1, h, 1);

    // 4) out = h @ W2 + b2
    gin_wmma_gemm<<<gemm_blocks, 128, 0, stream>>>(h, W2, b2, out, 0);
}